// DeepseekMoEGate_13262859010621
// MI455X (gfx1250) — compile-verified
//
#include <hip/hip_runtime.h>

typedef __attribute__((ext_vector_type(2))) float v2f;
typedef __attribute__((ext_vector_type(4))) float v4f;
typedef __attribute__((ext_vector_type(8))) float v8f;
typedef __attribute__((ext_vector_type(4))) int   v4i;

namespace {
constexpr int T   = 16384;   // tokens
constexpr int H   = 2048;    // hidden
constexpr int E   = 256;     // experts
constexpr int MT  = 64;      // tokens per workgroup
constexpr int KS  = 64;      // K elements staged per pipeline stage
constexpr int AST = 68;      // A LDS row stride (floats), padded vs 64 banks
constexpr int ABUF = MT * AST;          // floats per A buffer (4352)
constexpr int LST = 257;     // logits LDS row stride (floats)
constexpr int NBLK = T / MT; // 256 workgroups
constexpr int NSTAGE = H / KS;          // 32 pipeline stages
constexpr float NEG_INF = -3.402823466e38f;
}

// --- CDNA5 async global->LDS staging (ASYNCcnt path), with sync fallback ---
#if defined(__has_builtin)
#if __has_builtin(__builtin_amdgcn_global_load_async_to_lds_b128)
#define USE_ASYNC_LDS 1
#endif
#endif

#if defined(__has_builtin)
#if __has_builtin(__builtin_amdgcn_s_wait_asynccnt)
#define WAIT_ASYNC(N) __builtin_amdgcn_s_wait_asynccnt(N)
#endif
#endif
#ifndef WAIT_ASYNC
#define WAIT_ASYNC(N) asm volatile("s_wait_asynccnt " #N ::: "memory")
#endif

// Builtin prototype (per hipcc diagnostic): (v4i AS1*, v4i AS3*, imm, imm)
typedef __attribute__((address_space(1))) v4i* gv4i_p;
typedef __attribute__((address_space(3))) v4i* lv4i_p;

// Fused: gate GEMM (f32 WMMA 16x16x4) + grouped top-k + softmax + per-WG
// deterministic partial reductions for the aux/z losses.
__global__ __launch_bounds__(256) void moe_gate_fused(
    const float* __restrict__ X,   // [T, H]
    const float* __restrict__ W,   // [E, H]
    float* __restrict__ out,       // [T*8 idx | T*8 weight | 1 loss]
    float* __restrict__ pp,        // [NBLK, E] per-WG sum of softmax probs
    float* __restrict__ pc)        // [NBLK, E] per-WG selection counts
{
  extern __shared__ float smem[];
  float* Ab0  = smem;                      // [MT][AST] double-buffered A
  float* Ab1  = smem + ABUF;
  float* Lg   = smem;                      // [MT][LST] logits/probs after GEMM
  int*   cnt  = (int*)(smem + MT * LST);   // [E] selection counts

  const int tid  = threadIdx.x;
  const int wave = tid >> 5;      // 0..7 -> expert group
  const int lane = tid & 31;
  const int lh   = lane >> 4;     // lane half: selects K pair per ISA layout
  const int col  = lane & 15;     // M row (A) / N col (B,D)
  const int t0   = blockIdx.x * MT;

  if (tid < E) cnt[tid] = 0;

  // Hoisted per-thread B row pointers: inner loop uses immediate offsets only.
  const float* __restrict__ wp0 = W + (size_t)(wave * 32 +      col) * H + lh * 2;
  const float* __restrict__ wp1 = W + (size_t)(wave * 32 + 16 + col) * H + lh * 2;
  const int sidx0 = tid;          // chunk index base; 4 chunks: sidx0 + j*256

  v8f acc[4][2] = {};  // 4 M-tiles x 2 N-tiles of 16x16 f32

  auto stage_issue = [&](int k0s, float* Adst) {
    #pragma unroll
    for (int j = 0; j < (MT * KS / 4) / 256; j++) {
      int idx = j * 256 + sidx0;
      int row = idx >> 4;            // 16 float4 per row
      int c4  = (idx & 15) << 2;
      const float* src = X + (size_t)(t0 + row) * H + k0s + c4;
      float*       dst = Adst + row * AST + c4;
#ifdef USE_ASYNC_LDS
      __builtin_amdgcn_global_load_async_to_lds_b128(
          (gv4i_p)src, (lv4i_p)dst, 0, 0);
#else
      *(v4f*)dst = *(const v4f*)src;
      __builtin_prefetch(src + KS, 0, 1);
#endif
    }
  };

  stage_issue(0, Ab0);

  for (int i = 0; i < NSTAGE; i++) {
    const int k0 = i * KS;
    float* Acur = (i & 1) ? Ab1 : Ab0;
    float* Anxt = (i & 1) ? Ab0 : Ab1;

    __syncthreads();                       // all readers done with Anxt's old data
    if (i + 1 < NSTAGE) {
      stage_issue(k0 + KS, Anxt);          // overlap next HBM fetch with compute
      WAIT_ASYNC(4);                       // stage i landed; 4 newest still in flight
    } else {
      WAIT_ASYNC(0);
    }
    __syncthreads();                       // stage i visible to whole WG

    const float* arow = Acur + col * AST + lh * 2;
    const float* w0s  = wp0 + k0;
    const float* w1s  = wp1 + k0;
    #pragma unroll
    for (int s = 0; s < KS / 4; s++) {
      const int kc = s * 4;
      // B fragments (4x16): K = kc + lh*2 + {0,1}; immediate-offset loads.
      v2f b0 = *(const v2f*)(w0s + kc);
      v2f b1 = *(const v2f*)(w1s + kc);
      #pragma unroll
      for (int m = 0; m < 4; m++) {
        // A fragment (16x4): row M = col, K = kc + lh*2 + {0,1}
        v2f a = *(const v2f*)(arow + m * 16 * AST + kc);
        acc[m][0] = __builtin_amdgcn_wmma_f32_16x16x4_f32(
            false, a, false, b0, (short)0, acc[m][0], false, false);
        acc[m][1] = __builtin_amdgcn_wmma_f32_16x16x4_f32(
            false, a, false, b1, (short)0, acc[m][1], false, false);
      }
    }
  }
  __syncthreads();

  // Scatter accumulators into logits LDS. D layout: VGPR r -> M=r+lh*8, N=col.
  #pragma unroll
  for (int m = 0; m < 4; m++)
    #pragma unroll
    for (int n = 0; n < 2; n++)
      #pragma unroll
      for (int r = 0; r < 8; r++) {
        int tok = m * 16 + r + lh * 8;
        int ex  = wave * 32 + n * 16 + col;
        Lg[tok * LST + ex] = acc[m][n][r];
      }
  __syncthreads();

  // Per-token grouped top-k, softmax, full softmax (in place), counts.
  if (tid < MT) {
    float* row = &Lg[tid * LST];
    float cv[32]; int ci[32];
    for (int g = 0; g < 8; g++) {
      unsigned msk = 0;
      for (int p = 0; p < 4; p++) {
        float best = NEG_INF; int bi = 0;
        for (int i = 0; i < 32; i++) {
          if ((msk >> i) & 1u) continue;
          float v = row[g * 32 + i];
          if (v > best) { best = v; bi = i; }   // first-index-wins ties (jax)
        }
        msk |= 1u << bi;
        cv[g * 4 + p] = best; ci[g * 4 + p] = bi;
      }
    }
    unsigned m2 = 0; float tv[8]; int tix[8];
    for (int k = 0; k < 8; k++) {
      float best = NEG_INF; int bp = 0;
      for (int i = 0; i < 32; i++) {
        if ((m2 >> i) & 1u) continue;
        if (cv[i] > best) { best = cv[i]; bp = i; }
      }
      m2 |= 1u << bp;
      tv[k]  = best;
      tix[k] = (bp >> 2) * 32 + ci[bp];   // group*GROUP_SIZE + expert_in_group
    }
    float mx = tv[0], ssum = 0.f, w8[8];
    for (int k = 0; k < 8; k++) { w8[k] = __expf(tv[k] - mx); ssum += w8[k]; }
    float inv = 1.f / ssum;
    int gt = t0 + tid;
    for (int k = 0; k < 8; k++) {
      out[(size_t)gt * 8 + k]                 = (float)tix[k];
      out[(size_t)T * 8 + (size_t)gt * 8 + k] = w8[k] * inv;
    }
    // Full softmax over 256 experts, stored back in place for reduction.
    float fm = NEG_INF;
    for (int e2 = 0; e2 < E; e2++) fm = fmaxf(fm, row[e2]);
    float fs = 0.f;
    for (int e2 = 0; e2 < E; e2++) fs += __expf(row[e2] - fm);
    float inv2 = 1.f / fs;
    for (int e2 = 0; e2 < E; e2++) row[e2] = __expf(row[e2] - fm) * inv2;
    // Integer LDS atomics: exactly associative -> deterministic.
    for (int k = 0; k < 8; k++) atomicAdd(&cnt[tix[k]], 1);
  }
  __syncthreads();

  // Deterministic fixed-order column reduction -> per-WG partials (no float atomics).
  {
    int e2 = tid; float s = 0.f;
    for (int t = 0; t < MT; t++) s += Lg[t * LST + e2];
    pp[(size_t)blockIdx.x * E + e2] = s;
    pc[(size_t)blockIdx.x * E + e2] = (float)cnt[e2];
  }
}

__global__ __launch_bounds__(256) void moe_gate_finalize(
    const float* __restrict__ pp, const float* __restrict__ pc,
    float* __restrict__ out)
{
  __shared__ float ra[256], rz[256];
  int e = threadIdx.x;
  float S = 0.f, C = 0.f;
  for (int b = 0; b < NBLK; b++) {
    S += pp[(size_t)b * E + e];
    C += pc[(size_t)b * E + e];
  }
  float invT = 1.f / (float)T;
  ra[e] = (C * invT) * (S * invT);      // counts/T * expert_probs
  float lg = __logf(S);                 // log(sum_t probs)
  rz[e] = lg * lg;
  __syncthreads();
  for (int off = 128; off > 0; off >>= 1) {
    if (e < off) { ra[e] += ra[e + off]; rz[e] += rz[e + off]; }
    __syncthreads();
  }
  if (e == 0)
    out[(size_t)T * 16] = ra[0] * 0.001f + (rz[0] / (float)E) * 0.0001f;
}

extern "C" void kernel_launch(void* const* d_in, const int* in_sizes, int n_in,
                              void* d_out, int out_size, void* d_ws, size_t ws_size,
                              hipStream_t stream) {
  (void)in_sizes; (void)n_in; (void)out_size; (void)ws_size;
  const float* X = (const float*)d_in[0];
  const float* W = (const float*)d_in[1];
  float* out = (float*)d_out;
  float* pp = (float*)d_ws;                       // [NBLK*E]
  float* pc = pp + (size_t)NBLK * E;              // [NBLK*E]  (512 KB total)

  size_t smem = (size_t)(MT * LST) * sizeof(float) + E * sizeof(int);
  (void)hipFuncSetAttribute((const void*)moe_gate_fused,
                            hipFuncAttributeMaxDynamicSharedMemorySize,
                            (int)smem);
  moe_gate_fused<<<dim3(NBLK), dim3(256), smem, stream>>>(X, W, out, pp, pc);
  moe_gate_finalize<<<dim3(1), dim3(256), 0, stream>>>(pp, pc, out);
}